// SpanResolverModel_57492432224979
// MI455X (gfx1250) — compile-verified
//
#include <hip/hip_runtime.h>
#include <math.h>

// CDNA5 fp32 WMMA: D(16x16) = A(16x4) * B(4x16) + C
typedef float v2f __attribute__((ext_vector_type(2)));
typedef float v8f __attribute__((ext_vector_type(8)));

#define WMMA_F32(a, b, c) \
  __builtin_amdgcn_wmma_f32_16x16x4_f32(false, (a), false, (b), (short)0, (c), false, false)

// ---- async global->LDS path (guarded: falls back to load+ds_store if absent) ----
typedef int b128_t __attribute__((vector_size(16)));

#if defined(__gfx1250__) &&                                          \
    __has_builtin(__builtin_amdgcn_global_load_async_to_lds_b32) &&  \
    __has_builtin(__builtin_amdgcn_global_load_async_to_lds_b128) && \
    __has_builtin(__builtin_amdgcn_s_wait_asynccnt)
#define USE_ASYNC_LDS 1
#define GL_ASYNC_B32(gp, lp)                                        \
  __builtin_amdgcn_global_load_async_to_lds_b32(                    \
      (__attribute__((address_space(1))) int*)(gp),                 \
      (__attribute__((address_space(3))) int*)(lp), 0, 0)
#define GL_ASYNC_B128(gp, lp)                                       \
  __builtin_amdgcn_global_load_async_to_lds_b128(                   \
      (__attribute__((address_space(1))) b128_t*)(gp),              \
      (__attribute__((address_space(3))) b128_t*)(lp), 0, 0)
#define WAIT_ASYNC() __builtin_amdgcn_s_wait_asynccnt(0)
#else
#define USE_ASYNC_LDS 0
#endif

static __device__ inline v8f zero8() {
  v8f z = {0.f, 0.f, 0.f, 0.f, 0.f, 0.f, 0.f, 0.f};
  return z;
}

// ---------------------------------------------------------------------------
// Kernel 0: fill output with -inf (entire [H, W, 2] tensor)
// ---------------------------------------------------------------------------
__global__ __launch_bounds__(256) void fill_neg_inf(float4* __restrict__ out, size_t n4) {
  const float ninf = -__builtin_inff();
  float4 v = make_float4(ninf, ninf, ninf, ninf);
  size_t i = (size_t)blockIdx.x * blockDim.x + threadIdx.x;
  const size_t stride = (size_t)gridDim.x * blockDim.x;
  for (; i < n4; i += stride) out[i] = v;
}

// ---------------------------------------------------------------------------
// Generic fp32 WMMA GEMM:  C[M,N] = A[M,K] @ B[K,N] (+bias)
// Block tile 64x128, kt=32, 8 waves: wave w -> row-tile (w&3), col-tiles 4*(w>>2)..+3
// Requires: M%64==0, N%128==0, K%32==0.
// ---------------------------------------------------------------------------
__global__ __launch_bounds__(256) void gemm_f32_wmma(
    const float* __restrict__ A, int lda, const float* __restrict__ B, int ldb,
    float* __restrict__ C, int ldc, const float* __restrict__ bias, int K) {
  __shared__ float As[64 * 36];   // [64][36] row-major A tile (pitch 36: aligned + conflict-free)
  __shared__ float Bt[128 * 36];  // [128][36] transposed B tile: Bt[n][k]
  const int t = threadIdx.x;
  const int lane = t & 31, wv = t >> 5;
  const int rt = wv & 3, cg = wv >> 2;
  const int m0 = blockIdx.y * 64, n0 = blockIdx.x * 128;

  v8f acc[4];
#pragma unroll
  for (int i = 0; i < 4; i++) acc[i] = zero8();

  const int am = t >> 2, akq = (t & 3) * 8;  // A staging: row am, cols akq..akq+7
  const int bk = t >> 3, bnq = (t & 7) * 16; // B staging: k-row bk, 16 cols from bnq

  for (int k0 = 0; k0 < K; k0 += 32) {
    __syncthreads();
#if USE_ASYNC_LDS
    // A tile: straight strided copy, async b128 (per-lane LDS destinations)
    GL_ASYNC_B128(&A[(size_t)(m0 + am) * lda + k0 + akq], &As[am * 36 + akq]);
    GL_ASYNC_B128(&A[(size_t)(m0 + am) * lda + k0 + akq + 4], &As[am * 36 + akq + 4]);
    // B tile: transpose via per-lane-addressed async b32
#pragma unroll
    for (int j = 0; j < 16; j++) {
      const int g = (wv * 16 + j) * 32 + lane;  // 0..4095 over block
      const int k = g >> 7, n = g & 127;        // k:0..31, n:0..127
      GL_ASYNC_B32(&B[(size_t)(k0 + k) * ldb + n0 + n], &Bt[n * 36 + k]);
    }
    WAIT_ASYNC();
#else
    // stage A tile
    {
      const float4 a0 = *(const float4*)&A[(size_t)(m0 + am) * lda + k0 + akq];
      const float4 a1 = *(const float4*)&A[(size_t)(m0 + am) * lda + k0 + akq + 4];
      *(float4*)&As[am * 36 + akq] = a0;
      *(float4*)&As[am * 36 + akq + 4] = a1;
    }
    // stage B tile (transpose into LDS so (k,k+1) contiguous per column)
#pragma unroll
    for (int j = 0; j < 4; j++) {
      const int n = bnq + j * 4;
      const float4 bv = *(const float4*)&B[(size_t)(k0 + bk) * ldb + n0 + n];
      Bt[(n + 0) * 36 + bk] = bv.x;
      Bt[(n + 1) * 36 + bk] = bv.y;
      Bt[(n + 2) * 36 + bk] = bv.z;
      Bt[(n + 3) * 36 + bk] = bv.w;
    }
#endif
    __syncthreads();
#pragma unroll
    for (int ks = 0; ks < 8; ks++) {
      const int kk = ks * 4 + ((lane >> 4) << 1);
      const v2f af = *(const v2f*)&As[(rt * 16 + (lane & 15)) * 36 + kk];
#pragma unroll
      for (int i = 0; i < 4; i++) {
        const int n = (cg * 4 + i) * 16 + (lane & 15);
        const v2f bf = *(const v2f*)&Bt[n * 36 + kk];
        acc[i] = WMMA_F32(af, bf, acc[i]);
      }
    }
  }
  // epilogue: C/D layout — lane<16: rows rt*16+r; lane>=16: rows rt*16+8+r
#pragma unroll
  for (int i = 0; i < 4; i++) {
    const int n = (cg * 4 + i) * 16 + (lane & 15);
    const float bb = bias ? bias[n0 + n] : 0.0f;
#pragma unroll
    for (int r = 0; r < 8; r++) {
      const int m = rt * 16 + ((lane >> 4) << 3) + r;
      C[(size_t)(m0 + m) * ldc + n0 + n] = acc[i][r] + bb;
    }
  }
}

// ---------------------------------------------------------------------------
// Fused per-head kernel: 2 heads per block (M = 64 packed rows).
// Phase 1: x1 = relu(P0[head]+P1[col]+Pe[eid])  -> WMMA x2 = relu(x1 @ w2 + b2)
// Phase 2: x3 = x2 @ w3 + b3 (WMMA)
// Phase 3: conv1 (64ch,k3,pad1) -> conv2 (2ch,k3,pad1) in VALU
// Phase 4: masked scatter into [H, 4096, 2] with start/end validity
// LDS: one 64KB buffer, regions reused across barriers.
// ---------------------------------------------------------------------------
__global__ __launch_bounds__(256) void span_head_kernel(
    const float* __restrict__ P0, const float* __restrict__ P1,
    const float* __restrict__ Pe, const int* __restrict__ heads_ids,
    const float* __restrict__ w2, const float* __restrict__ b2,
    const float* __restrict__ w3, const float* __restrict__ b3,
    const float* __restrict__ c1w, const float* __restrict__ c1b,
    const float* __restrict__ c2w, const float* __restrict__ c2b,
    float* __restrict__ out) {
  __shared__ float smem[16384];  // 64KB
  float* As = smem;              // phase1: x1 tile [64][36]
  float* Bt = smem + 64 * 36;    // phase1: w2^T tile [256][36]

  const int t = threadIdx.x;
  const int lane = t & 31, wv = t >> 5;
  const int rt = wv & 3, ng = wv >> 2;  // row-tile 0..3, n-group 0..1 (128 cols each)
  const int h0 = blockIdx.x * 2;

  // per-thread x1 gather rows (fixed across k-blocks)
  const int am = t >> 2;          // packed row 0..63
  const int akq = (t & 3) * 8;    // k sub-offset 0/8/16/24
  const int hlA = am >> 5, pA = am & 31;
  const int hwA = heads_ids[h0 + hlA];
  const int colA = ((hwA >> 5) << 5) + pA;
  const int eidA = hwA - colA + 63;  // always in [32,94]
  const float* p0row = P0 + (size_t)hwA * 1024 + akq;
  const float* p1row = P1 + (size_t)colA * 1024 + akq;
  const float* perow = Pe + (size_t)eidA * 1024 + akq;

  const int bk = t >> 3;          // 0..31 (k row within tile)
  const int bnq = (t & 7) * 32;   // 32 cols per thread

  v8f acc[8];
#pragma unroll
  for (int i = 0; i < 8; i++) acc[i] = zero8();

  // ---- Phase 1: K=1024 loop, kt=32: x2 = relu(x1 @ w2 + b2) ----
  for (int kb = 0; kb < 32; kb++) {
    __syncthreads();
#if USE_ASYNC_LDS
    // stage w2^T tile via per-lane-addressed async b32 (transpose in the async engine)
#pragma unroll
    for (int j = 0; j < 32; j++) {
      const int g = (wv * 32 + j) * 32 + lane;  // 0..8191 over block
      const int k = g >> 8, n = g & 255;        // k:0..31, n:0..255
      GL_ASYNC_B32(&w2[(size_t)kb * 8192 + g], &Bt[n * 36 + k]);
    }
#endif
    {  // build x1 tile: relu(P0 + P1 + Pe)
      const int kg = kb * 32;
      const float4 s0 = *(const float4*)&p0row[kg];
      const float4 s1 = *(const float4*)&p1row[kg];
      const float4 s2 = *(const float4*)&perow[kg];
      const float4 t0 = *(const float4*)&p0row[kg + 4];
      const float4 t1 = *(const float4*)&p1row[kg + 4];
      const float4 t2 = *(const float4*)&perow[kg + 4];
      float4 r0, r1;
      r0.x = fmaxf(s0.x + s1.x + s2.x, 0.f);
      r0.y = fmaxf(s0.y + s1.y + s2.y, 0.f);
      r0.z = fmaxf(s0.z + s1.z + s2.z, 0.f);
      r0.w = fmaxf(s0.w + s1.w + s2.w, 0.f);
      r1.x = fmaxf(t0.x + t1.x + t2.x, 0.f);
      r1.y = fmaxf(t0.y + t1.y + t2.y, 0.f);
      r1.z = fmaxf(t0.z + t1.z + t2.z, 0.f);
      r1.w = fmaxf(t0.w + t1.w + t2.w, 0.f);
      *(float4*)&As[am * 36 + akq] = r0;
      *(float4*)&As[am * 36 + akq + 4] = r1;
    }
#if USE_ASYNC_LDS
    WAIT_ASYNC();
#else
    // stage w2^T tile (VGPR round-trip fallback)
#pragma unroll
    for (int j = 0; j < 8; j++) {
      const int n = bnq + j * 4;
      const float4 bv = *(const float4*)&w2[(size_t)(kb * 32 + bk) * 256 + n];
      Bt[(n + 0) * 36 + bk] = bv.x;
      Bt[(n + 1) * 36 + bk] = bv.y;
      Bt[(n + 2) * 36 + bk] = bv.z;
      Bt[(n + 3) * 36 + bk] = bv.w;
    }
#endif
    if (kb < 31)  // prefetch next w2 tile (global_prefetch_b8)
      __builtin_prefetch(&w2[(size_t)((kb + 1) * 32 + bk) * 256 + bnq], 0, 1);
    __syncthreads();
#pragma unroll
    for (int ks = 0; ks < 8; ks++) {
      const int kk = ks * 4 + ((lane >> 4) << 1);
      const v2f af = *(const v2f*)&As[(rt * 16 + (lane & 15)) * 36 + kk];
#pragma unroll
      for (int i = 0; i < 8; i++) {
        const v2f bf = *(const v2f*)&Bt[(ng * 128 + i * 16 + (lane & 15)) * 36 + kk];
        acc[i] = WMMA_F32(af, bf, acc[i]);
      }
    }
  }
  __syncthreads();
  // write x2 = relu(acc + b2) to smem as [64][256] (fills entire buffer)
#pragma unroll
  for (int i = 0; i < 8; i++) {
    const int n = ng * 128 + i * 16 + (lane & 15);
    const float bb = b2[n];
#pragma unroll
    for (int r = 0; r < 8; r++) {
      const int m = rt * 16 + ((lane >> 4) << 3) + r;
      smem[m * 256 + n] = fmaxf(acc[i][r] + bb, 0.f);
    }
  }
  __syncthreads();

  // ---- Phase 2: x3 = x2 @ w3 + b3  (M=64, N=64, K=256) ----
  v8f a3[2];
  a3[0] = zero8();
  a3[1] = zero8();
#pragma unroll 4
  for (int ks = 0; ks < 64; ks++) {
    const int kk = ks * 4 + ((lane >> 4) << 1);
    const v2f af = *(const v2f*)&smem[(rt * 16 + (lane & 15)) * 256 + kk];
#pragma unroll
    for (int i = 0; i < 2; i++) {
      const int n = (ng * 2 + i) * 16 + (lane & 15);
      v2f bf;
      bf.x = w3[(size_t)kk * 64 + n];
      bf.y = w3[(size_t)(kk + 1) * 64 + n];
      a3[i] = WMMA_F32(af, bf, a3[i]);
    }
  }
  __syncthreads();  // all x2 reads done; buffer reusable
  // x3 -> smem[0..4096) as [64][64]
#pragma unroll
  for (int i = 0; i < 2; i++) {
    const int n = (ng * 2 + i) * 16 + (lane & 15);
    const float bb = b3[n];
#pragma unroll
    for (int r = 0; r < 8; r++) {
      const int m = rt * 16 + ((lane >> 4) << 3) + r;
      smem[m * 64 + n] = a3[i][r] + bb;
    }
  }
  __syncthreads();

  // ---- Phase 3a: conv1 (in=64ch, out=64ch, k=3, pad=1) over packed axis ----
  float* y1 = smem + 4096;  // [2][64][32]
#pragma unroll
  for (int it = 0; it < 16; it++) {
    const int id = t + it * 256;  // 0..4095
    const int hl = id >> 11, c = (id >> 5) & 63, p = id & 31;
    float sum = c1b[c];
#pragma unroll
    for (int k = 0; k < 3; k++) {
      const int pp = p + k - 1;
      if (pp >= 0 && pp < 32) {
        const float* xr = &smem[(hl * 32 + pp) * 64];
        const float* wr = &c1w[(size_t)c * 192 + k];
        for (int ii = 0; ii < 64; ii++) sum += xr[ii] * wr[ii * 3];
      }
    }
    y1[(hl * 64 + c) * 32 + p] = sum;
  }
  __syncthreads();

  // ---- Phase 3b: conv2 (in=64ch, out=2ch, k=3, pad=1) ----
  float* y2 = smem + 8192;  // [2][2][32]
  if (t < 128) {
    const int hl = t >> 6, o = (t >> 5) & 1, p = t & 31;
    float sum = c2b[o];
    for (int c = 0; c < 64; c++) {
#pragma unroll
      for (int k = 0; k < 3; k++) {
        const int pp = p + k - 1;
        if (pp >= 0 && pp < 32)
          sum += c2w[(size_t)(o * 64 + c) * 3 + k] * y1[(hl * 64 + c) * 32 + pp];
      }
    }
    y2[(hl * 2 + o) * 32 + p] = sum;
  }
  __syncthreads();

  // ---- Phase 4: masked scatter ----
  if (t < 128) {
    const int hl = t >> 6, p = (t >> 1) & 31, o = t & 1;
    const int h = h0 + hl;
    const int hw = heads_ids[h];
    const int w = ((hw >> 5) << 5) + p;
    const int rel = hw - w;
    const bool ok = (o == 0) ? (rel >= 0) : (rel <= 0);
    out[((size_t)h * 4096 + w) * 2 + o] = ok ? y2[(hl * 2 + o) * 32 + p] : -__builtin_inff();
  }
}

// ---------------------------------------------------------------------------
extern "C" void kernel_launch(void* const* d_in, const int* in_sizes, int n_in,
                              void* d_out, int out_size, void* d_ws, size_t ws_size,
                              hipStream_t stream) {
  (void)in_sizes; (void)n_in; (void)out_size; (void)ws_size;
  const float* words = (const float*)d_in[1];   // [4096, 768]
  const int* heads_ids = (const int*)d_in[2];   // [2048]
  const float* emb = (const float*)d_in[3];     // [128, 64]
  const float* w1 = (const float*)d_in[4];      // [1600, 1024]
  const float* b1 = (const float*)d_in[5];      // [1024]
  const float* w2 = (const float*)d_in[6];      // [1024, 256]
  const float* b2 = (const float*)d_in[7];      // [256]
  const float* w3 = (const float*)d_in[8];      // [256, 64]
  const float* b3 = (const float*)d_in[9];      // [64]
  const float* c1w = (const float*)d_in[10];    // [64, 64, 3]
  const float* c1b = (const float*)d_in[11];    // [64]
  const float* c2w = (const float*)d_in[12];    // [2, 64, 3]
  const float* c2b = (const float*)d_in[13];    // [2]
  float* out = (float*)d_out;                   // [2048, 4096, 2]

  float* P0 = (float*)d_ws;                     // [4096, 1024]
  float* P1 = P0 + (size_t)4096 * 1024;         // [4096, 1024]
  float* Pe = P1 + (size_t)4096 * 1024;         // [128, 1024]

  // 1) init output to -inf
  fill_neg_inf<<<4096, 256, 0, stream>>>((float4*)out, (size_t)2048 * 4096 * 2 / 4);
  // 2) word projections through w1's head-rows and word-rows
  gemm_f32_wmma<<<dim3(8, 64), 256, 0, stream>>>(words, 768, w1, 1024, P0, 1024, nullptr, 768);
  gemm_f32_wmma<<<dim3(8, 64), 256, 0, stream>>>(words, 768, w1 + (size_t)768 * 1024, 1024, P1, 1024, nullptr, 768);
  // 3) distance-embedding projection (+b1 folded here)
  gemm_f32_wmma<<<dim3(8, 2), 256, 0, stream>>>(emb, 64, w1 + (size_t)1536 * 1024, 1024, Pe, 1024, b1, 64);
  // 4) fused gather + FFNN (WMMA) + conv stack + masked scatter
  span_head_kernel<<<1024, 256, 0, stream>>>(P0, P1, Pe, heads_ids, w2, b2, w3, b3,
                                             c1w, c1b, c2w, c2b, out);
}